// LSTMModel_46420006535830
// MI455X (gfx1250) — compile-verified
//
#include <hip/hip_runtime.h>
#include <math.h>

#define T_SEQ   16384
#define HID     800
#define NGATE   3200
#define NWG     200        // workgroups in the persistent scan
#define UPW     4          // hidden units per WG  (NWG*UPW == HID)
#define ROWS    16         // gate rows per WG     (4 gates * UPW)
#define BLOCK   256        // 8 wave32 per WG

typedef float v2f __attribute__((ext_vector_type(2)));
typedef float v8f __attribute__((ext_vector_type(8)));

// ---- workspace layout (floats) ----
// [0]                : grid-barrier counter (as uint32)
// [16 .. 16+1600)    : h double buffer [2][HID]
// [1616 .. 2416)     : h_last [HID]
// [2416 .. 3416)     : t1 [1000]  (MLP hidden)
#define WS_HBUF   16
#define WS_HLAST  (16 + 2*HID)
#define WS_T1     (16 + 2*HID + HID)

__device__ __forceinline__ float sigf(float x) { return 1.0f / (1.0f + expf(-x)); }

__global__ __launch_bounds__(BLOCK, 1) void init_ws_kernel(float* ws) {
    for (int i = threadIdx.x; i < WS_HBUF + 2 * HID; i += BLOCK) ws[i] = 0.0f;
}

__global__ __launch_bounds__(BLOCK, 1) void lstm_scan_kernel(
    const float* __restrict__ input_seq,   // [T_SEQ, 20]
    const float* __restrict__ W_ih,        // [3200, 20]
    const float* __restrict__ W_hh,        // [3200, 800]
    const float* __restrict__ b_ih,        // [3200]
    const float* __restrict__ b_hh,        // [3200]
    float* __restrict__ ws)
{
    __shared__ float w_s[ROWS][HID];   // this WG's slice of W_hh, LDS-resident
    __shared__ float h_s[HID];         // staged h_{t-1}
    __shared__ float xp_s[16][16];     // WMMA-produced x_proj tile (16 t x 16 rows)
    __shared__ float gate_s[ROWS];
    __shared__ float c_s[UPW];

    const int tid  = threadIdx.x;
    const int lane = tid & 31;
    const int wave = tid >> 5;
    const int j0   = blockIdx.x * UPW;           // first hidden unit owned

    unsigned* counter = (unsigned*)ws;
    float* hbuf  = ws + WS_HBUF;                 // [2][HID]
    float* hlast = ws + WS_HLAST;

    // ---- load this WG's 16 gate rows of W_hh into LDS (once) ----
    for (int r = 0; r < ROWS; ++r) {
        const int grow = (r >> 2) * HID + j0 + (r & 3);   // gate block * H + unit
        for (int k = tid; k < HID; k += BLOCK)
            w_s[r][k] = W_hh[(size_t)grow * HID + k];
    }
    if (tid < UPW) c_s[tid] = 0.0f;
    __syncthreads();

    for (int t = 0; t < T_SEQ; ++t) {
        // ---- phase A: stage h_{t-1}; every 16 steps wave0 builds xp tile via WMMA
        for (int k = tid; k < HID; k += BLOCK)
            h_s[k] = hbuf[(t & 1) * HID + k];

        if (wave == 0 && (t & 15) == 0) {
            const int t0    = t;
            const int nloc  = lane & 15;      // N (local gate row) / M (local t)
            const int khalf = lane >> 4;      // K-half select per 16x16x4 layout
            const int grow  = (nloc >> 2) * HID + j0 + (nloc & 3);

            v8f c;
            const float bias = b_ih[grow] + b_hh[grow];
            #pragma unroll
            for (int v = 0; v < 8; ++v) c[v] = bias;     // bias folded into accumulator

            #pragma unroll
            for (int k0 = 0; k0 < 20; k0 += 4) {         // K=20 -> 5 x v_wmma_f32_16x16x4_f32
                const int k = k0 + 2 * khalf;
                v2f a = *(const v2f*)(input_seq + (size_t)(t0 + nloc) * 20 + k); // A: X tile
                v2f b = *(const v2f*)(W_ih      + (size_t)grow        * 20 + k); // B: W_ih^T
                c = __builtin_amdgcn_wmma_f32_16x16x4_f32(
                        false, a, false, b, (short)0, c, false, false);
            }
            #pragma unroll
            for (int v = 0; v < 8; ++v)
                xp_s[v + 8 * khalf][nloc] = c[v];        // D layout: M = v + 8*(lane>=16)

            if (t0 + 16 < T_SEQ)                          // hint next tile (global_prefetch_b8)
                __builtin_prefetch(input_seq + (size_t)(t0 + 16) * 20 + lane * 10, 0, 0);
        }
        __syncthreads();

        // ---- phase B: 16 dot products of length 800 from LDS-resident weights
        {
            const int r0 = wave * 2, r1 = r0 + 1;
            float a0 = 0.0f, a1 = 0.0f;
            #pragma unroll
            for (int i = 0; i < 25; ++i) {               // 800 = 32 lanes * 25
                const int k = lane + 32 * i;
                const float hv = h_s[k];
                a0 = fmaf(w_s[r0][k], hv, a0);
                a1 = fmaf(w_s[r1][k], hv, a1);
            }
            #pragma unroll
            for (int off = 16; off > 0; off >>= 1) {
                a0 += __shfl_xor(a0, off, 32);
                a1 += __shfl_xor(a1, off, 32);
            }
            if (lane == 0) {
                gate_s[r0] = a0 + xp_s[t & 15][r0];
                gate_s[r1] = a1 + xp_s[t & 15][r1];
            }
        }
        __syncthreads();

        // ---- phase C: gate nonlinearity + state update (lanes 0..3 of wave 0)
        if (tid < UPW) {
            const float iv = sigf(gate_s[tid]);
            const float fv = sigf(gate_s[4 + tid]);
            const float gv = tanhf(gate_s[8 + tid]);
            const float ov = sigf(gate_s[12 + tid]);
            const float cn = fv * c_s[tid] + iv * gv;
            c_s[tid] = cn;
            const float hn = ov * tanhf(cn);
            hbuf[((t + 1) & 1) * HID + j0 + tid] = hn;   // double-buffered broadcast
            if (t == T_SEQ - 1) hlast[j0 + tid] = hn;
        }

        // ---- phase D: device-wide split barrier (monotonic counter, no reset)
        if (tid == 0) {
            __threadfence();                              // h stores (same wave) -> visible
            __hip_atomic_fetch_add(counter, 1u, __ATOMIC_RELEASE, __HIP_MEMORY_SCOPE_AGENT);
            const unsigned target = (unsigned)(t + 1) * NWG;
            while (__hip_atomic_load(counter, __ATOMIC_ACQUIRE, __HIP_MEMORY_SCOPE_AGENT) < target)
                __builtin_amdgcn_s_sleep(1);
        }
        __syncthreads();
    }
}

// t1[m] = b1[m] + sum_k h_last[k] * W1[m,k]   (one wave per output row, coalesced)
__global__ __launch_bounds__(BLOCK, 1) void mlp1_kernel(
    const float* __restrict__ W1, const float* __restrict__ b1, float* __restrict__ ws)
{
    const int lane = threadIdx.x & 31;
    const int m = blockIdx.x * 8 + (threadIdx.x >> 5);   // 125 blocks * 8 waves = 1000
    const float* h = ws + WS_HLAST;
    float acc = 0.0f;
    #pragma unroll
    for (int i = 0; i < 25; ++i) {
        const int k = lane + 32 * i;
        acc = fmaf(W1[(size_t)m * HID + k], h[k], acc);
    }
    #pragma unroll
    for (int off = 16; off > 0; off >>= 1) acc += __shfl_xor(acc, off, 32);
    if (lane == 0) ws[WS_T1 + m] = acc + b1[m];
}

// out[o] = b2[o] + sum_m t1[m] * W2[o,m]
__global__ __launch_bounds__(BLOCK, 1) void mlp2_kernel(
    const float* __restrict__ W2, const float* __restrict__ b2,
    const float* __restrict__ ws, float* __restrict__ out)
{
    const int lane = threadIdx.x & 31;
    const int wave = threadIdx.x >> 5;
    const float* t1 = ws + WS_T1;
    for (int o = wave; o < 20; o += 8) {
        float acc = 0.0f;
        for (int k = lane; k < 1000; k += 32)
            acc = fmaf(W2[(size_t)o * 1000 + k], t1[k], acc);
        #pragma unroll
        for (int off = 16; off > 0; off >>= 1) acc += __shfl_xor(acc, off, 32);
        if (lane == 0) out[o] = acc + b2[o];
    }
}

extern "C" void kernel_launch(void* const* d_in, const int* in_sizes, int n_in,
                              void* d_out, int out_size, void* d_ws, size_t ws_size,
                              hipStream_t stream) {
    const float* input_seq = (const float*)d_in[0];
    const float* W_ih      = (const float*)d_in[1];
    const float* W_hh      = (const float*)d_in[2];
    const float* b_ih      = (const float*)d_in[3];
    const float* b_hh      = (const float*)d_in[4];
    const float* W1        = (const float*)d_in[5];
    const float* b1        = (const float*)d_in[6];
    const float* W2        = (const float*)d_in[7];
    const float* b2        = (const float*)d_in[8];
    float* out = (float*)d_out;
    float* ws  = (float*)d_ws;
    (void)in_sizes; (void)n_in; (void)out_size; (void)ws_size;

    init_ws_kernel<<<1, BLOCK, 0, stream>>>(ws);
    lstm_scan_kernel<<<NWG, BLOCK, 0, stream>>>(input_seq, W_ih, W_hh, b_ih, b_hh, ws);
    mlp1_kernel<<<125, BLOCK, 0, stream>>>(W1, b1, ws);
    mlp2_kernel<<<1, BLOCK, 0, stream>>>(W2, b2, ws, out);
}